// StaticAttention_90709709292093
// MI455X (gfx1250) — compile-verified
//
#include <hip/hip_runtime.h>
#include <math.h>

#define B_   128
#define S_   256
#define D_   256
#define H_   8
#define HD_  2048
#define EPS_ 1e-5f

typedef __attribute__((ext_vector_type(2))) float v2f;
typedef __attribute__((ext_vector_type(8))) float v8f;

// ---- workspace layout (float offsets), total ~2.1 MB ----
#define WS_GK     0          // 256x256 Gram of Wk
#define WS_GV     65536      // 256x256 Gram of Wv
#define WS_WKSUM  131072     // colsum(Wk) [256]
#define WS_HK     131328     // Wk @ bk    [256]
#define WS_WVSUM  131584
#define WS_HV     131840
#define WS_SC     132096     // [bksum, |bk|^2, bvsum, |bv|^2]
#define WS_Q      132352     // q = LN(local@Wq+bq) [128*2048]
#define WS_MUK    394496     // per-row LN mean of kraw  [128*256]
#define WS_INVK   427264     // per-row LN inv-std       [128*256]
#define WS_MUV    460032
#define WS_INVV   492800

// LDS bank padding for k_stats (64 banks x 4B):
//  - G tile stride 264 (== 8 mod 64): B-operand fetch (16 consecutive cols,
//    two K-halves 2 rows apart) touches 32 distinct banks.
//  - A tile stride 260 (== 4 mod 64): A-operand fetch (16 rows, pair of K)
//    lands on banks {4*l16 + k}: conflict-free for both lane halves.
#define GSTRIDE 264
#define ASTRIDE 260

__device__ __forceinline__ float blk_reduce(float v, float* red) {
  const int tid = threadIdx.x;
  red[tid] = v;
  __syncthreads();
  #pragma unroll
  for (int off = 128; off > 0; off >>= 1) {
    if (tid < off) red[tid] += red[tid + off];
    __syncthreads();
  }
  float r = red[0];
  __syncthreads();
  return r;
}

__device__ __forceinline__ float wv_sum(float v) {
  #pragma unroll
  for (int o = 16; o > 0; o >>= 1) v += __shfl_xor(v, o, 32);
  return v;
}
__device__ __forceinline__ float wv_max(float v) {
  #pragma unroll
  for (int o = 16; o > 0; o >>= 1) v = fmaxf(v, __shfl_xor(v, o, 32));
  return v;
}

// ---------- fixed reductions of Wk/Wv and biases ----------
__global__ void k_prep(const float* __restrict__ Wk, const float* __restrict__ bk,
                       const float* __restrict__ Wv, const float* __restrict__ bv,
                       float* __restrict__ ws) {
  const int mat = blockIdx.x;
  const float* W    = mat ? Wv : Wk;
  const float* bias = mat ? bv : bk;
  __shared__ float lb[HD_];
  __shared__ float red[256];
  const int tid = threadIdx.x;
  for (int j = tid; j < HD_; j += 256) lb[j] = bias[j];
  __syncthreads();
  float s1 = 0.f, s2 = 0.f;
  const float* row = W + (size_t)tid * HD_;
  for (int j = 0; j < HD_; ++j) { float w = row[j]; s1 += w; s2 += w * lb[j]; }
  ws[(mat ? WS_WVSUM : WS_WKSUM) + tid] = s1;
  ws[(mat ? WS_HV : WS_HK) + tid] = s2;
  float pb = 0.f, pb2 = 0.f;
  for (int j = tid; j < HD_; j += 256) { float x = lb[j]; pb += x; pb2 += x * x; }
  float bsum = blk_reduce(pb, red);
  float b2   = blk_reduce(pb2, red);
  if (tid == 0) { ws[WS_SC + mat * 2 + 0] = bsum; ws[WS_SC + mat * 2 + 1] = b2; }
}

// ---------- Gram matrices G = W * W^T (256x2048x256), f32 WMMA ----------
__global__ void k_gram(const float* __restrict__ Wk, const float* __restrict__ Wv,
                       float* __restrict__ ws) {
  const int mat = blockIdx.y;
  const float* W = mat ? Wv : Wk;
  float* G = ws + (mat ? WS_GV : WS_GK);
  const int tid  = threadIdx.x;
  const int lane = tid & 31, wave = tid >> 5;
  const int l16 = lane & 15, hf = lane >> 4;
  const int tile = blockIdx.x * 8 + wave;           // 256 tiles / matrix
  const int m0 = (tile >> 4) << 4, n0 = (tile & 15) << 4;
  const float* arow = W + (size_t)(m0 + l16) * HD_ + 2 * hf;
  const float* brow = W + (size_t)(n0 + l16) * HD_ + 2 * hf;  // B[k,n] = W[n,k]
  v8f acc = {};
  for (int k0 = 0; k0 < HD_; k0 += 4) {
    v2f a, b;
    a.x = arow[k0]; a.y = arow[k0 + 1];
    b.x = brow[k0]; b.y = brow[k0 + 1];
    acc = __builtin_amdgcn_wmma_f32_16x16x4_f32(false, a, false, b, (short)0, acc,
                                                false, false);
  }
  #pragma unroll
  for (int i = 0; i < 8; ++i)
    G[(m0 + i + 8 * hf) * 256 + n0 + l16] = acc[i];
}

// ---------- q = LN(local @ Wq + bq) ----------
__global__ void k_qln(const float* __restrict__ local, const float* __restrict__ Wq,
                      const float* __restrict__ bq, const float* __restrict__ gq,
                      const float* __restrict__ betq, float* __restrict__ ws) {
  const int b = blockIdx.x, tid = threadIdx.x;
  __shared__ float xl[D_];
  __shared__ float red[256];
  xl[tid] = local[b * D_ + tid];
  __syncthreads();
  float acc[8];
  #pragma unroll
  for (int i = 0; i < 8; ++i) {
    int hd = tid + 256 * i;
    float s = bq[hd];
    for (int c = 0; c < D_; ++c) s += xl[c] * Wq[(size_t)c * HD_ + hd];
    acc[i] = s;
  }
  float ps = 0.f, pq = 0.f;
  #pragma unroll
  for (int i = 0; i < 8; ++i) { ps += acc[i]; pq += acc[i] * acc[i]; }
  float mean = blk_reduce(ps, red) * (1.f / HD_);
  float e2   = blk_reduce(pq, red) * (1.f / HD_);
  float inv  = rsqrtf(e2 - mean * mean + EPS_);
  #pragma unroll
  for (int i = 0; i < 8; ++i) {
    int hd = tid + 256 * i;
    ws[WS_Q + b * HD_ + hd] = (acc[i] - mean) * inv * gq[hd] + betq[hd];
  }
}

// ---------- per-row LN stats of kraw/vraw via quadratic form x^T G x ----------
// LDS-resident: full Gram matrix (bank-padded) + per-m-tile A slab feed the
// f32 WMMA pipe from ds_load; global traffic per block ~= X once + G once.
__global__ void __launch_bounds__(256) k_stats(const float* __restrict__ dist,
                                               const float* __restrict__ sites,
                                               float* __restrict__ ws) {
  const int b = blockIdx.x, mat = blockIdx.y, tid = threadIdx.x;
  const float* X = (mat ? sites : dist) + (size_t)b * S_ * D_;   // 256x256
  const float* G = ws + (mat ? WS_GV : WS_GK);
  __shared__ float Gl[256 * GSTRIDE];   // 264 KiB (padded)
  __shared__ float At[16 * ASTRIDE];    // 16.3 KiB (padded)
  __shared__ float lws[D_], lh[D_], mu[S_], dh[S_], rowsum[S_];
  lws[tid] = ws[(mat ? WS_WVSUM : WS_WKSUM) + tid];
  lh[tid]  = ws[(mat ? WS_HV : WS_HK) + tid];
  rowsum[tid] = 0.f;
  // stage G into LDS (coalesced global reads)
  for (int idx = tid; idx < 256 * 256; idx += 256)
    Gl[(idx >> 8) * GSTRIDE + (idx & 255)] = G[idx];
  __syncthreads();
  {  // mean and x.(W b)
    const float* xr = X + (size_t)tid * D_;
    float d1 = 0.f, d2 = 0.f;
    for (int c = 0; c < D_; ++c) { float x = xr[c]; d1 += x * lws[c]; d2 += x * lh[c]; }
    mu[tid] = (d1 + ws[WS_SC + mat * 2]) * (1.f / HD_);
    dh[tid] = d2;
  }
  __syncthreads();
  // Y = X @ G via WMMA f32; accumulate diag(Y X^T) per row
  const int lane = tid & 31, wave = tid >> 5;
  const int l16 = lane & 15, hf = lane >> 4;
  for (int mt = 0; mt < 16; ++mt) {
    const int m0 = mt * 16;
    // cooperative stage of A slab: rows m0..m0+15 of X
    for (int idx = tid; idx < 16 * 256; idx += 256)
      At[(idx >> 8) * ASTRIDE + (idx & 255)] = X[(size_t)(m0 + (idx >> 8)) * D_ + (idx & 255)];
    __syncthreads();
    float dacc[8] = {0.f, 0.f, 0.f, 0.f, 0.f, 0.f, 0.f, 0.f};
    const float* ar = At + l16 * ASTRIDE + 2 * hf;
    for (int nt = wave; nt < 16; nt += 8) {
      const int n0 = nt * 16;
      v8f acc = {};
      for (int k0 = 0; k0 < D_; k0 += 4) {
        v2f a, bb;
        a.x = ar[k0]; a.y = ar[k0 + 1];
        bb.x = Gl[(k0 + 2 * hf) * GSTRIDE + n0 + l16];
        bb.y = Gl[(k0 + 2 * hf + 1) * GSTRIDE + n0 + l16];
        acc = __builtin_amdgcn_wmma_f32_16x16x4_f32(false, a, false, bb, (short)0, acc,
                                                    false, false);
      }
      #pragma unroll
      for (int i = 0; i < 8; ++i)
        dacc[i] += acc[i] * At[(i + 8 * hf) * ASTRIDE + n0 + l16];
    }
    // combine the 16 lanes sharing each row, then one LDS atomic per row/wave
    #pragma unroll
    for (int i = 0; i < 8; ++i) {
      float v = dacc[i];
      v += __shfl_xor(v, 1, 32); v += __shfl_xor(v, 2, 32);
      v += __shfl_xor(v, 4, 32); v += __shfl_xor(v, 8, 32);
      if (l16 == 0) atomicAdd(&rowsum[m0 + i + 8 * hf], v);
    }
    __syncthreads();  // protect At before next stage, rowsum before phase 2
  }
  {
    float b2 = ws[WS_SC + mat * 2 + 1];
    float e2 = (rowsum[tid] + 2.f * dh[tid] + b2) * (1.f / HD_);
    float m  = mu[tid];
    float inv = rsqrtf(e2 - m * m + EPS_);
    ws[(mat ? WS_MUV : WS_MUK) + b * S_ + tid]   = m;
    ws[(mat ? WS_INVV : WS_INVK) + b * S_ + tid] = inv;
  }
}

// ---------- fused attention + output head (one block per batch) ----------
__global__ void k_attn(const float* __restrict__ dist, const float* __restrict__ sites,
                       const unsigned char* __restrict__ mask,
                       const float* __restrict__ Wk, const float* __restrict__ bk,
                       const float* __restrict__ gk, const float* __restrict__ betk,
                       const float* __restrict__ Wv, const float* __restrict__ bv,
                       const float* __restrict__ gv, const float* __restrict__ betv,
                       const float* __restrict__ g1, const float* __restrict__ b1,
                       const float* __restrict__ Wo, const float* __restrict__ bo,
                       const float* __restrict__ g2, const float* __restrict__ b2,
                       const float* __restrict__ beta_a,
                       const float* __restrict__ ws, float* __restrict__ out) {
  const int b = blockIdx.x, tid = threadIdx.x;
  const int lane = tid & 31, wave = tid >> 5;  // wave == head
  __shared__ float lq[HD_], lqg[HD_], wf[HD_], uat[HD_];
  __shared__ float red[256];
  __shared__ float hsA[H_], hscb[H_], hsBt[H_], hsT1[H_], hsT2[H_];

  #pragma unroll
  for (int i = 0; i < 8; ++i) {
    int j = tid + 256 * i;
    float q = ws[WS_Q + b * HD_ + j];
    lq[j] = q; lqg[j] = q * gk[j];
  }
  __syncthreads();
  {  // per-head scalars
    const int h = wave;
    float A = 0.f, cb = 0.f, Bt = 0.f;
    for (int d = lane; d < D_; d += 32) {
      int hd = h * D_ + d;
      float qg = lqg[hd];
      A += qg; cb += qg * bk[hd]; Bt += lq[hd] * betk[hd];
    }
    A = wv_sum(A); cb = wv_sum(cb); Bt = wv_sum(Bt);
    if (lane == 0) { hsA[h] = A; hscb[h] = cb; hsBt[h] = Bt; }
  }
  __syncthreads();
  {  // fold Wk against q*gk: wf[h*256+c]
    const int c = tid;
    const float* wrow = Wk + (size_t)c * HD_;
    #pragma unroll
    for (int h = 0; h < H_; ++h) {
      float s = 0.f;
      for (int d = 0; d < D_; ++d) s += wrow[h * D_ + d] * lqg[h * D_ + d];
      wf[h * D_ + c] = s;
    }
  }
  __syncthreads();
  {  // logits u[h,s]
    const int s = tid;
    float dot[H_];
    #pragma unroll
    for (int h = 0; h < H_; ++h) dot[h] = 0.f;
    const float* xr = dist + ((size_t)b * S_ + s) * D_;
    for (int c = 0; c < D_; ++c) {
      float x = xr[c];
      #pragma unroll
      for (int h = 0; h < H_; ++h) dot[h] += x * wf[h * D_ + c];
    }
    float muk = ws[WS_MUK + b * S_ + s], ik = ws[WS_INVK + b * S_ + s];
    bool mk = mask[b * S_ + s] != 0;
    #pragma unroll
    for (int h = 0; h < H_; ++h) {
      float u = 0.0625f * (ik * (dot[h] + hscb[h] - muk * hsA[h]) + hsBt[h]);
      uat[h * S_ + s] = mk ? -INFINITY : u;
    }
  }
  __syncthreads();
  {  // masked softmax per head; store c_s = attn*inv_v in-place
    const int h = wave;
    float v[8], mx = -INFINITY;
    #pragma unroll
    for (int k = 0; k < 8; ++k) { v[k] = uat[h * S_ + lane + 32 * k]; mx = fmaxf(mx, v[k]); }
    mx = wv_max(mx);
    float sm = 0.f;
    #pragma unroll
    for (int k = 0; k < 8; ++k) { v[k] = expf(v[k] - mx); sm += v[k]; }
    sm = wv_sum(sm);
    float isum = 1.f / sm, T1 = 0.f, T2 = 0.f;
    #pragma unroll
    for (int k = 0; k < 8; ++k) {
      int s = lane + 32 * k;
      float cs = v[k] * isum * ws[WS_INVV + b * S_ + s];
      T1 += cs; T2 += cs * ws[WS_MUV + b * S_ + s];
      uat[h * S_ + s] = cs;
    }
    T1 = wv_sum(T1); T2 = wv_sum(T2);
    if (lane == 0) { hsT1[h] = T1; hsT2[h] = T2; }
  }
  __syncthreads();
  {  // t_h = sum_s c_s * sites_s  (reuse wf)
    const int h = wave;
    float t[8];
    #pragma unroll
    for (int k = 0; k < 8; ++k) t[k] = 0.f;
    for (int s = 0; s < S_; ++s) {
      float cs = uat[h * S_ + s];
      const float* sr = sites + ((size_t)b * S_ + s) * D_;
      #pragma unroll
      for (int k = 0; k < 8; ++k) t[k] += cs * sr[lane + 32 * k];
    }
    #pragma unroll
    for (int k = 0; k < 8; ++k) wf[h * D_ + lane + 32 * k] = t[k];
  }
  __syncthreads();
  {  // out_attn + residual (reuse lqg as out buffer)
    const int h = wave;
    float T1 = hsT1[h], T2 = hsT2[h];
    #pragma unroll
    for (int k = 0; k < 8; ++k) {
      int d = lane + 32 * k, hd = h * D_ + d;
      float o = 0.f;
      for (int c = 0; c < D_; ++c) o += wf[h * D_ + c] * Wv[(size_t)c * HD_ + hd];
      lqg[hd] = gv[hd] * (o + T1 * bv[hd] - T2) + betv[hd] + lq[hd];
    }
  }
  __syncthreads();
  {  // LN1 over 2048, in place
    float ps = 0.f, pq = 0.f;
    #pragma unroll
    for (int i = 0; i < 8; ++i) { float x = lqg[tid + 256 * i]; ps += x; pq += x * x; }
    float mean = blk_reduce(ps, red) * (1.f / HD_);
    float e2   = blk_reduce(pq, red) * (1.f / HD_);
    float inv  = rsqrtf(e2 - mean * mean + EPS_);
    #pragma unroll
    for (int i = 0; i < 8; ++i) {
      int j = tid + 256 * i;
      lqg[j] = (lqg[j] - mean) * inv * g1[j] + b1[j];
    }
  }
  __syncthreads();
  {  // y = z @ Wo + bo, then LN2 over 256; plus trivial 'a' output (== beta_a)
    const int j = tid;
    float acc = bo[j];
    for (int hd = 0; hd < HD_; ++hd) acc += lqg[hd] * Wo[(size_t)hd * D_ + j];
    float mean = blk_reduce(acc, red) * (1.f / D_);
    float e2   = blk_reduce(acc * acc, red) * (1.f / D_);
    float inv  = rsqrtf(e2 - mean * mean + EPS_);
    out[32768 + b * D_ + j] = (acc - mean) * inv * g2[j] + b2[j];
    out[b * S_ + tid] = beta_a[0];  // LN over a length-1 axis is exactly beta
  }
}

extern "C" void kernel_launch(void* const* d_in, const int* in_sizes, int n_in,
                              void* d_out, int out_size, void* d_ws, size_t ws_size,
                              hipStream_t stream) {
  (void)in_sizes; (void)n_in; (void)out_size; (void)ws_size;
  const float* local = (const float*)d_in[0];
  const float* dist  = (const float*)d_in[1];
  const float* sites = (const float*)d_in[2];
  const unsigned char* mask = (const unsigned char*)d_in[3];
  const float* Wq   = (const float*)d_in[4];
  const float* bq   = (const float*)d_in[5];
  const float* gq   = (const float*)d_in[6];
  const float* betq = (const float*)d_in[7];
  const float* Wk   = (const float*)d_in[8];
  const float* bk   = (const float*)d_in[9];
  const float* gk   = (const float*)d_in[10];
  const float* betk = (const float*)d_in[11];
  const float* Wv   = (const float*)d_in[12];
  const float* bv   = (const float*)d_in[13];
  const float* gv   = (const float*)d_in[14];
  const float* betv = (const float*)d_in[15];
  const float* g1   = (const float*)d_in[16];
  const float* b1   = (const float*)d_in[17];
  const float* Wo   = (const float*)d_in[18];
  const float* bo   = (const float*)d_in[19];
  const float* g2   = (const float*)d_in[20];
  const float* b2   = (const float*)d_in[21];
  // d_in[22..24] = Wa, ba, ga: algebraically eliminated (LN over size-1 axis)
  const float* beta_a = (const float*)d_in[25];
  float* ws  = (float*)d_ws;
  float* out = (float*)d_out;

  k_prep <<<2, 256, 0, stream>>>(Wk, bk, Wv, bv, ws);
  k_gram <<<dim3(32, 2), 256, 0, stream>>>(Wk, Wv, ws);
  k_qln  <<<128, 256, 0, stream>>>(local, Wq, bq, gq, betq, ws);
  k_stats<<<dim3(128, 2), 256, 0, stream>>>(dist, sites, ws);
  k_attn <<<128, 256, 0, stream>>>(dist, sites, mask, Wk, bk, gk, betk,
                                   Wv, bv, gv, betv, g1, b1, Wo, bo, g2, b2,
                                   beta_a, ws, out);
}